// CenterIdLoss_77309412134
// MI455X (gfx1250) — compile-verified
//
#include <hip/hip_runtime.h>
#include <hip/hip_bf16.h>
#include <math.h>

#define N_SAMPLES 8192
#define C_DIM     4096
// scale = NUM_POS / N^2 = 4 / 8192^2 = 1/16777216

typedef __attribute__((ext_vector_type(2))) float v2f;
typedef __attribute__((ext_vector_type(8))) float v8f;

// ---------------- init: zero counts, cursor, losses, out ----------------
__global__ void k_init(int* __restrict__ counts, int* __restrict__ cursor,
                       float* __restrict__ losses, float* __restrict__ out) {
    int i = blockIdx.x * blockDim.x + threadIdx.x;
    if (i < C_DIM) {
        counts[i] = 0;
        cursor[i] = 0;
        losses[i] = 0.0f;
    }
    if (i == 0) out[0] = 0.0f;
}

// ---------------- histogram of labels ----------------
__global__ void k_hist(const int* __restrict__ label, int* __restrict__ counts) {
    int i = blockIdx.x * blockDim.x + threadIdx.x;
    if (i < N_SAMPLES) atomicAdd(&counts[label[i]], 1);
}

// ---------------- exclusive scan of counts (single block, 1024 thr) ----------------
__global__ void k_scan(const int* __restrict__ counts, int* __restrict__ offsets) {
    __shared__ int sc[1024];
    int t = threadIdx.x;
    int c0 = counts[4 * t + 0];
    int c1 = counts[4 * t + 1];
    int c2 = counts[4 * t + 2];
    int c3 = counts[4 * t + 3];
    int tot = c0 + c1 + c2 + c3;
    sc[t] = tot;
    __syncthreads();
    for (int off = 1; off < 1024; off <<= 1) {
        int v = (t >= off) ? sc[t - off] : 0;
        __syncthreads();
        sc[t] += v;
        __syncthreads();
    }
    int excl = sc[t] - tot;  // exclusive prefix of 4-element groups
    offsets[4 * t + 0] = excl;
    offsets[4 * t + 1] = excl + c0;
    offsets[4 * t + 2] = excl + c0 + c1;
    offsets[4 * t + 3] = excl + c0 + c1 + c2;
}

// ---------------- scatter sample ids into per-class buckets ----------------
__global__ void k_scatter(const int* __restrict__ label, const int* __restrict__ offsets,
                          int* __restrict__ cursor, int* __restrict__ sorted) {
    int i = blockIdx.x * blockDim.x + threadIdx.x;
    if (i < N_SAMPLES) {
        int c = label[i];
        int p = atomicAdd(&cursor[c], 1);
        sorted[offsets[c] + p] = i;
    }
}

// ---------------- main: one workgroup per class ----------------
// LDS layout (floats): rb[4][4112] row stage (16-dword pad => conflict-free B loads),
// ctr[4096] center-row column sums, red[256] reduce scratch, sAccC[1].
#define RB_STRIDE 4112

__launch_bounds__(256)
__global__ void k_class(const float* __restrict__ feat, const int* __restrict__ counts,
                        const int* __restrict__ offsets, const int* __restrict__ sorted,
                        float* __restrict__ losses) {
    const int cls = blockIdx.x;
    const int cnt = counts[cls];
    if (cnt == 0) return;  // empty classes contribute nothing (losses pre-zeroed)

    const int base = offsets[cls];
    const int tid  = threadIdx.x;
    const int lane = tid & 31;
    const int wave = tid >> 5;

    extern __shared__ float smem[];
    float* rb    = smem;                      // 4 * RB_STRIDE
    float* ctr   = smem + 4 * RB_STRIDE;      // C_DIM
    float* red   = ctr + C_DIM;               // 256
    float* sAccC = red + 256;                 // 1

    if (tid == 0) sAccC[0] = 0.0f;
    // zero ctr once -> chunk loop does unconditional accumulate (no ch==0 branch)
    for (int j = tid; j < C_DIM; j += 256) ctr[j] = 0.0f;

    const float inv_cnt = 1.0f / (float)cnt;
    const int nChunks = (cnt + 3) >> 2;

    const int half = lane >> 4;   // 0: lanes 0-15, 1: lanes 16-31
    const int n    = lane & 15;
    const int k0   = half * 2;    // B VGPR0 -> row k0, VGPR1 -> row k0+1

    for (int ch = 0; ch < nChunks; ++ch) {
        int m = cnt - ch * 4;
        if (m > 4) m = 4;

        // zero-fill invalid staged rows so the all-ones WMMA column sums add 0 for them
        for (int r = m; r < 4; ++r)
            for (int j = tid; j < C_DIM; j += 256) rb[r * RB_STRIDE + j] = 0.0f;

        // async-stage m rows: coalesced B128 global -> LDS (CDNA5 async path)
        for (int r = 0; r < m; ++r) {
            int srcRow = sorted[base + ch * 4 + r];
            unsigned long long saddr =
                (unsigned long long)(feat + (size_t)srcRow * C_DIM);
            unsigned ldsBase =
                (unsigned)(unsigned long long)(rb + r * RB_STRIDE);
#pragma unroll
            for (int j = 0; j < 4; ++j) {
                unsigned off = (unsigned)((tid + 256 * j) * 16);  // byte offset in row
                unsigned lds = ldsBase + off;
                asm volatile("global_load_async_to_lds_b128 %0, %1, %2"
                             :
                             : "v"(lds), "v"(off), "s"(saddr)
                             : "memory");
            }
        }
        asm volatile("s_wait_asynccnt 0" ::: "memory");
        __syncthreads();

        // exact (column-correct) accumulation of column `cls`
        if (tid == 0) {
            float a = sAccC[0];
            for (int r = 0; r < m; ++r) a += rb[r * RB_STRIDE + cls];
            sAccC[0] = a;
        }

        // WMMA f32 16x16x4 with all-ones A: D[m][n] = sum_k B[k][n].
        // Two 16-col tiles per iteration; lanes 0-15 commit tile A, lanes 16-31
        // commit tile B (all D rows identical) -> divergence-free coalesced store.
        v2f a1;
        a1.x = 1.0f;
        a1.y = 1.0f;
#pragma unroll 1
        for (int t = 0; t < 16; ++t) {
            int b0 = wave * 512 + t * 32;  // base column of the tile pair
            v2f bb0, bb1;
            bb0.x = rb[(k0 + 0) * RB_STRIDE + b0 + n];
            bb0.y = rb[(k0 + 1) * RB_STRIDE + b0 + n];
            bb1.x = rb[(k0 + 0) * RB_STRIDE + b0 + 16 + n];
            bb1.y = rb[(k0 + 1) * RB_STRIDE + b0 + 16 + n];
            v8f d0 = {};
            v8f d1 = {};
            d0 = __builtin_amdgcn_wmma_f32_16x16x4_f32(
                     false, a1, false, bb0, (short)0, d0, false, false);
            d1 = __builtin_amdgcn_wmma_f32_16x16x4_f32(
                     false, a1, false, bb1, (short)0, d1, false, false);
            float val = (lane < 16) ? d0[0] : d1[0];
            ctr[b0 + lane] += val;
        }
        __syncthreads();
    }

    // logsumexp over center row = ctr * inv_cnt  (permutation-invariant)
    float lmax = -INFINITY;
    for (int j = tid; j < C_DIM; j += 256) {
        float v = ctr[j] * inv_cnt;
        lmax = fmaxf(lmax, v);
    }
    red[tid] = lmax;
    __syncthreads();
    for (int s = 128; s > 0; s >>= 1) {
        if (tid < s) red[tid] = fmaxf(red[tid], red[tid + s]);
        __syncthreads();
    }
    float gmax = red[0];
    __syncthreads();

    float lsum = 0.0f;
    for (int j = tid; j < C_DIM; j += 256) {
        float v = ctr[j] * inv_cnt;
        lsum += expf(v - gmax);
    }
    red[tid] = lsum;
    __syncthreads();
    for (int s = 128; s > 0; s >>= 1) {
        if (tid < s) red[tid] += red[tid + s];
        __syncthreads();
    }

    if (tid == 0) {
        float lse     = logf(red[0]) + gmax;
        float centerC = sAccC[0] * inv_cnt;
        losses[cls]   = (float)cnt * (lse - centerC);
    }
}

// ---------------- deterministic final reduction ----------------
__global__ void k_final(const float* __restrict__ losses, float* __restrict__ out) {
    __shared__ float red[256];
    int tid = threadIdx.x;
    float s = 0.0f;
    for (int j = tid; j < C_DIM; j += 256) s += losses[j];
    red[tid] = s;
    __syncthreads();
    for (int st = 128; st > 0; st >>= 1) {
        if (tid < st) red[tid] += red[tid + st];
        __syncthreads();
    }
    if (tid == 0) out[0] = red[0] * (1.0f / 16777216.0f);  // * NUM_POS / N^2
}

extern "C" void kernel_launch(void* const* d_in, const int* in_sizes, int n_in,
                              void* d_out, int out_size, void* d_ws, size_t ws_size,
                              hipStream_t stream) {
    const float* feat  = (const float*)d_in[0];
    const int*   label = (const int*)d_in[1];
    float*       out   = (float*)d_out;

    // workspace layout
    int*   counts  = (int*)d_ws;
    int*   offsets = counts + C_DIM;
    int*   cursor  = offsets + C_DIM;
    int*   sorted  = cursor + C_DIM;
    float* losses  = (float*)(sorted + N_SAMPLES);

    k_init<<<(C_DIM + 255) / 256, 256, 0, stream>>>(counts, cursor, losses, out);
    k_hist<<<N_SAMPLES / 256, 256, 0, stream>>>(label, counts);
    k_scan<<<1, 1024, 0, stream>>>(counts, offsets);
    k_scatter<<<N_SAMPLES / 256, 256, 0, stream>>>(label, offsets, cursor, sorted);

    size_t shmem = (size_t)(4 * RB_STRIDE + C_DIM + 256 + 4) * sizeof(float);
    k_class<<<C_DIM, 256, shmem, stream>>>(feat, counts, offsets, sorted, losses);

    k_final<<<1, 256, 0, stream>>>(losses, out);
}